// ModuleSepconv_72198400246186
// MI455X (gfx1250) — compile-verified
//
#include <hip/hip_runtime.h>
#include <stdint.h>

// Problem constants (match reference)
#define FN 51
#define HOUT 512
#define WOUT 512
#define BN 2
#define CN 3
#define HIN (HOUT + FN - 1)   // 562
#define WIN (WOUT + FN - 1)   // 562

#define KPAD 52               // F padded to multiple of 4 for 16x16x4 K-steps
#define TSTRIP 66             // 16 output x + 50 extra input columns
#define SCOLS 80              // strip padded to 5 WMMA N-tiles (cols 66..79 stay 0)
#define GSTRIDE 81            // G row stride (bank-conflict-free diagonal reads)
#define SBUF (KPAD * SCOLS)   // 4160 floats per strip buffer (rows contiguous)

typedef float v2f __attribute__((ext_vector_type(2)));
typedef float v8f __attribute__((ext_vector_type(8)));

// Flat shared-memory VA keeps the wave-relative LDS offset in its low 32 bits
// (ISA 10.2, LDS aperture).
__device__ __forceinline__ uint32_t lds_off(const void* p) {
  return (uint32_t)(uintptr_t)p;
}
// GLOBAL_LOAD_ASYNC_TO_LDS (GVS form): saddr = uniform 64-bit base,
// vaddr = per-lane unsigned byte offset, %0 = per-lane LDS byte address.
__device__ __forceinline__ void async_b64(uint32_t lds, uint32_t voff, uint64_t base) {
  asm volatile("global_load_async_to_lds_b64 %0, %1, %2"
               :: "v"(lds), "v"(voff), "s"(base) : "memory");
}
__device__ __forceinline__ void async_b128(uint32_t lds, uint32_t voff, uint64_t base) {
  asm volatile("global_load_async_to_lds_b128 %0, %1, %2"
               :: "v"(lds), "v"(voff), "s"(base) : "memory");
}
__device__ __forceinline__ void wait_async0() {
  asm volatile("s_wait_asynccnt 0x0" ::: "memory");
}

// Stage one 51x66 input strip: row walk, lane owns t = 2*lane (t=0..63),
// constant-increment addressing (1 async + 2 adds per row), tiny tail for t=64.
__device__ __forceinline__ void stage_strip(float* buf, const float* irow, int lane) {
  const uint64_t base = (uint64_t)(uintptr_t)irow;
  uint32_t voff = (uint32_t)(lane * 8);             // 2 floats per lane
  uint32_t loff = lds_off(buf) + (uint32_t)(lane * 8);
  #pragma unroll 3
  for (int i = 0; i < FN; ++i) {
    async_b64(loff, voff, base);
    voff += WIN * 4;                                // +2248 B (8B-aligned rows)
    loff += SCOLS * 4;                              // +320 B
  }
  for (int e = lane; e < FN; e += 32)               // leftover pair t = 64,65
    async_b64(lds_off(&buf[e * SCOLS + 64]), (uint32_t)((e * WIN + 64) * 4), base);
}

__global__ __launch_bounds__(32) void sepconv_wmma_kernel(
    const float* __restrict__ inp, const float* __restrict__ vert,
    const float* __restrict__ horiz, float* __restrict__ out)
{
  __shared__ float sV[KPAD * 16];    // vertical kernels, [i][n], row 51 = 0
  __shared__ float sH[KPAD * 16];    // horizontal kernels, [j][n], row 51 = 0
  __shared__ float sS[2 * SBUF];     // double-buffered input strip, pads = 0
  __shared__ float sG[16 * GSTRIDE]; // G = V^T * S, [n][t]

  const int lane = threadIdx.x;      // 0..31 (wave32)
  const int x0 = blockIdx.x * 16;
  const int y  = blockIdx.y;
  const int b  = blockIdx.z;
  const int PS = HOUT * WOUT;        // kernel-plane stride (elements)

  const float* vrow = vert  + (size_t)b * FN * PS + (size_t)y * WOUT + x0;
  const float* hrow = horiz + (size_t)b * FN * PS + (size_t)y * WOUT + x0;

  // ---- zero ONLY pad cells (disjoint from async-written cells -> no race) --
  if (lane < 16) { sV[FN * 16 + lane] = 0.f; sH[FN * 16 + lane] = 0.f; }
  // cols 66..79 of all 104 strip rows (both buffers are row-contiguous)
  for (int e = lane; e < 2 * KPAD * 16; e += 32) {
    const int r = e >> 4, tt = e & 15;
    if (tt < 14) sS[r * SCOLS + 66 + tt] = 0.f;
  }
  // row 51 of each buffer, cols 0..65
  for (int e = lane; e < 2 * TSTRIP; e += 32) {
    const int bi = e >= TSTRIP;
    sS[(FN + bi * KPAD) * SCOLS + (e - (bi ? TSTRIP : 0))] = 0.f;
  }

  // ---- async-stage V and H tiles (51 rows x 16 floats, 16B-aligned b128) ---
  {
    const uint64_t vb = (uint64_t)(uintptr_t)vrow;
    const uint64_t hb = (uint64_t)(uintptr_t)hrow;
    for (int e = lane; e < FN * 4; e += 32) {        // 204 chunks of 4 floats
      const int i = e >> 2, q = (e & 3) * 4;
      const uint32_t go = (uint32_t)((i * PS + q) * 4);  // shifts only
      async_b128(lds_off(&sV[i * 16 + q]), go, vb);
      async_b128(lds_off(&sH[i * 16 + q]), go, hb);
    }
  }
  // ---- async-stage channel 0 input strip ----------------------------------
  stage_strip(&sS[0], inp + ((size_t)(b * CN) * HIN + y) * WIN + x0, lane);
  wait_async0();
  __syncthreads();

  // ---- Build A fragments (16x4 f32 per ISA layout), shared by all channels -
  // lanes 0-15: M=lane, VGPR0=K0, VGPR1=K1 ; lanes 16-31: M=lane-16, K2/K3
  v2f Afrag[13];
  {
    const int m  = lane & 15;
    const int kh = (lane >> 4) << 1;
    #pragma unroll
    for (int kk = 0; kk < 13; ++kk) {
      const int i0 = kk * 4 + kh;                    // kk=12,kh=2 -> zero row 51
      Afrag[kk].x = sV[i0 * 16 + m];
      Afrag[kk].y = sV[(i0 + 1) * 16 + m];
    }
  }

  for (int c = 0; c < CN; ++c) {
    const int cur = c & 1;
    // ---- prefetch next channel's strip into the other buffer (overlapped) --
    if (c + 1 < CN)
      stage_strip(&sS[(1 - cur) * SBUF],
                  inp + ((size_t)(b * CN + c + 1) * HIN + y) * WIN + x0, lane);

    // ---- Stage 1: G(16x80) = V^T(16x51) * S(51x80) via v_wmma_f32_16x16x4 --
    const float* buf = &sS[cur * SBUF];
    const int tn = lane & 15;
    const int kh = (lane >> 4) << 1;
    #pragma unroll
    for (int nt = 0; nt < 5; ++nt) {
      const int t0 = nt * 16;
      v8f acc;
      #pragma unroll
      for (int g = 0; g < 8; ++g) acc[g] = 0.f;
      #pragma unroll
      for (int kk = 0; kk < 13; ++kk) {
        const int i0 = kk * 4 + kh;
        v2f Bfrag;                         // B(4x16): N in lanes, K split like A
        Bfrag.x = buf[i0 * SCOLS + t0 + tn];
        Bfrag.y = buf[(i0 + 1) * SCOLS + t0 + tn];
        acc = __builtin_amdgcn_wmma_f32_16x16x4_f32(
            false, Afrag[kk], false, Bfrag, (short)0, acc, false, false);
      }
      // D layout: VGPR g -> row g (lanes 0-15) / row 8+g (lanes 16-31)
      const int row0 = (lane >> 4) * 8;
      #pragma unroll
      for (int g = 0; g < 8; ++g)
        sG[(row0 + g) * GSTRIDE + t0 + tn] = acc[g];
    }
    __syncthreads();

    // ---- Stage 2: out[n] = sum_j h[j,n] * G[n, n+j]; taps split across halves
    const int n  = lane & 15;
    const int jb = (lane >> 4) * 26;       // lanes 0-15: j 0..25 ; 16-31: 26..51
    float p = 0.f;
    #pragma unroll 2
    for (int jj = 0; jj < 26; ++jj) {
      const int j = jb + jj;               // j==51 hits zeroed sH row / zero G col
      p = __builtin_fmaf(sH[j * 16 + n], sG[n * GSTRIDE + n + j], p);
    }
    p += __shfl_down(p, 16, 32);           // combine the two half-wave partials
    if (lane < 16)
      out[((size_t)(b * CN + c) * HOUT + y) * WOUT + x0 + n] = p;

    wait_async0();                         // next buffer fully landed in LDS
    __syncthreads();                       // before next channel reuses sG
  }
}

extern "C" void kernel_launch(void* const* d_in, const int* in_sizes, int n_in,
                              void* d_out, int out_size, void* d_ws, size_t ws_size,
                              hipStream_t stream) {
  const float* inp   = (const float*)d_in[0];   // (B, C, 562, 562) f32
  const float* vert  = (const float*)d_in[1];   // (B, 51, 512, 512) f32
  const float* horiz = (const float*)d_in[2];   // (B, 51, 512, 512) f32
  float* out = (float*)d_out;                   // (B, C, 512, 512) f32
  dim3 grid(WOUT / 16, HOUT, BN);               // 32 x 512 x 2 tiles
  sepconv_wmma_kernel<<<grid, 32, 0, stream>>>(inp, vert, horiz, out);
}